// MixedDimLinearEmbedding_43705587204389
// MI455X (gfx1250) — compile-verified
//
#include <hip/hip_runtime.h>
#include <hip/hip_bf16.h>

typedef float v2f __attribute__((ext_vector_type(2)));
typedef float v8f __attribute__((ext_vector_type(8)));

#define DIM_HEAD 64
#define DIM_MID  32
#define DIM_TAIL 16
#define UD       64
#define KCHUNKS  12   // 8 mid (K=32) + 4 tail (K=16) -> fused K=48 GEMM

// One wave32 processes 16 batch rows.
// Fused GEMM: A[16x48] = [E_mid | E_tail], out-of-group sections zeroed via
// branch-free cndmask AFTER unconditional loads through index-clamped pointers
// (keeps all 12 A loads in flight -> partial s_wait_loadcnt, no exec churn).
// B[48x64] = [W_mid^T ; W_tail^T] hoisted to registers; C seeded w/ row bias.
// 48x v_wmma_f32_16x16x4_f32 per tile. Head rows: unconditional clamped gather
// + cndmask select. NT stores/index loads keep ~90MB of tables L2-resident
// while 512MB of output streams through.
__global__ __launch_bounds__(256) void mixed_embed_wmma_kernel(
    const int*   __restrict__ x,
    const int*   __restrict__ grp,
    const float* __restrict__ head,
    const float* __restrict__ mid,
    const float* __restrict__ tail,
    const float* __restrict__ Wm,   // [64,32] row-major
    const float* __restrict__ bm,   // [64]
    const float* __restrict__ Wt,   // [64,16] row-major
    const float* __restrict__ bt,   // [64]
    float*       __restrict__ out,  // [B,64]
    int batch)
{
    const int lane = threadIdx.x & 31;
    const int lo   = lane & 15;   // row (A) / col (B,C,D) within tile
    const int hi   = lane >> 4;   // K-pair selector / +8 row offset in C/D

    const int wavesPerBlock = blockDim.x >> 5;
    const int wave   = blockIdx.x * wavesPerBlock + (threadIdx.x >> 5);
    const int nwaves = gridDim.x * wavesPerBlock;

    // ---- Hoist fused B = [W_mid^T ; W_tail^T] and biases into registers ----
    v2f Bc[4][KCHUNKS];
    float cbm[4], cbt[4];
#pragma unroll
    for (int n = 0; n < 4; ++n) {
        const int j = 16 * n + lo;            // output column
#pragma unroll
        for (int k0 = 0; k0 < 8; ++k0)
            Bc[n][k0] = *(const v2f*)(Wm + j * DIM_MID + 4 * k0 + 2 * hi);
#pragma unroll
        for (int k0 = 0; k0 < 4; ++k0)
            Bc[n][8 + k0] = *(const v2f*)(Wt + j * DIM_TAIL + 4 * k0 + 2 * hi);
        cbm[n] = bm[j];
        cbt[n] = bt[j];
    }

    const int tiles = (batch + 15) >> 4;
    for (int t = wave; t < tiles; t += nwaves) {   // wave-uniform loop
        const int row0 = t << 4;
        const int r    = row0 + lo;                // lanes 16-31 mirror 0-15
        const bool rv  = (r < batch);
        const int gl   = rv ? __builtin_nontemporal_load(grp + r) : -1;
        const int il   = rv ? __builtin_nontemporal_load(x + r)   : 0;

        // Index-clamped row pointers: always in-bounds, row 0 when not in group.
        const float* midrow  = mid  + (size_t)((gl == 1) ? il : 0) * DIM_MID;
        const float* tailrow = tail + (size_t)((gl == 2) ? il : 0) * DIM_TAIL;

        // ---- issue ALL A-operand loads unconditionally (12 in flight) ----
        v2f am[8], at[4];
#pragma unroll
        for (int k0 = 0; k0 < 8; ++k0)
            am[k0] = *(const v2f*)(midrow + 4 * k0 + 2 * hi);
#pragma unroll
        for (int k0 = 0; k0 < 4; ++k0)
            at[k0] = *(const v2f*)(tailrow + 4 * k0 + 2 * hi);

        // Per-output-row group/index (row m = v + 8*hi lives in lane m's copy).
        int gmv[8], imv[8];
#pragma unroll
        for (int v = 0; v < 8; ++v) {
            const int m = v + 8 * hi;
            gmv[v] = __shfl(gl, m, 32);
            imv[v] = __shfl(il, m, 32);
        }

        // Accumulators seeded with the per-row bias (0 for head / other rows).
        v8f Dc[4];
#pragma unroll
        for (int n = 0; n < 4; ++n) {
#pragma unroll
            for (int v = 0; v < 8; ++v)
                Dc[n][v] = (gmv[v] == 1) ? cbm[n]
                         : (gmv[v] == 2) ? cbt[n] : 0.0f;
        }

        // ---- branch-free zeroing of out-of-group A sections (cndmask) ----
        const bool pm = (gl == 1), pt = (gl == 2);
#pragma unroll
        for (int k0 = 0; k0 < 8; ++k0) {
            am[k0].x = pm ? am[k0].x : 0.0f;
            am[k0].y = pm ? am[k0].y : 0.0f;
        }
#pragma unroll
        for (int k0 = 0; k0 < 4; ++k0) {
            at[k0].x = pt ? at[k0].x : 0.0f;
            at[k0].y = pt ? at[k0].y : 0.0f;
        }

        // ---- fused projection: A[16x48] @ B[48x64], 48 WMMAs ----
#pragma unroll
        for (int k0 = 0; k0 < KCHUNKS; ++k0) {
            const v2f a = (k0 < 8) ? am[k0] : at[k0 - 8];
#pragma unroll
            for (int n = 0; n < 4; ++n)
                Dc[n] = __builtin_amdgcn_wmma_f32_16x16x4_f32(
                    false, a, false, Bc[n][k0], (short)0, Dc[n], false, false);
        }

        // ---- epilogue: unconditional clamped head gather (32 loads in flight)
        float hv[8][4];
#pragma unroll
        for (int v = 0; v < 8; ++v) {
            const float* hrow =
                head + (size_t)((gmv[v] == 0) ? imv[v] : 0) * DIM_HEAD;
#pragma unroll
            for (int n = 0; n < 4; ++n)
                hv[v][n] = hrow[16 * n + lo];
        }

        // ---- branch-free select + NT streaming store ----
#pragma unroll
        for (int v = 0; v < 8; ++v) {
            const int rm = row0 + v + 8 * hi;
            if (rm < batch) {
#pragma unroll
                for (int n = 0; n < 4; ++n) {
                    const float val = (gmv[v] == 0) ? hv[v][n] : Dc[n][v];
                    __builtin_nontemporal_store(
                        val, out + (size_t)rm * UD + 16 * n + lo);
                }
            }
        }
    }
}

extern "C" void kernel_launch(void* const* d_in, const int* in_sizes, int n_in,
                              void* d_out, int out_size, void* d_ws, size_t ws_size,
                              hipStream_t stream) {
    const int*   x    = (const int*)  d_in[0];
    const int*   grp  = (const int*)  d_in[1];
    const float* head = (const float*)d_in[2];
    const float* mid  = (const float*)d_in[3];
    const float* tail = (const float*)d_in[4];
    const float* Wm   = (const float*)d_in[5];
    const float* bm   = (const float*)d_in[6];
    const float* Wt   = (const float*)d_in[7];
    const float* bt   = (const float*)d_in[8];
    float* out = (float*)d_out;

    const int batch = in_sizes[0];
    const int threads = 256;                    // 8 waves/block on wave32
    const int wavesPerBlock = threads / 32;
    const int tiles = (batch + 15) / 16;
    int blocks = (tiles + wavesPerBlock - 1) / wavesPerBlock;
    if (blocks > 1024) blocks = 1024;           // grid-stride; amortize B preload
    if (blocks < 1) blocks = 1;

    hipLaunchKernelGGL(mixed_embed_wmma_kernel, dim3(blocks), dim3(threads), 0, stream,
                       x, grp, head, mid, tail, Wm, bm, Wt, bt, out, batch);
}